// ImplicitRHMCSampler_17300128269079
// MI455X (gfx1250) — compile-verified
//
#include <hip/hip_runtime.h>
#include <hip/hip_bf16.h>

#define LSTEPS 6
#define NFX    8
#define GAMMA  0.01f
#define LDPAD  4

typedef __attribute__((ext_vector_type(2))) float v2f;
typedef __attribute__((ext_vector_type(4))) float v4f;
typedef __attribute__((ext_vector_type(8))) float v8f;

__device__ __forceinline__ float softplus_f(float x) {
  // stable: max(x,0) + log1p(exp(-|x|))
  return fmaxf(x, 0.0f) + __logf(1.0f + __expf(-fabsf(x)));
}
__device__ __forceinline__ float sigmoid_f(float x) {
  return 1.0f / (1.0f + __expf(-x));
}

enum { AM_RAW = 0, AM_T = 1, AM_U = 2 };
enum { EPI_BIAS = 0, EPI_AXPY = 1, EPI_ZD = 2, EPI_ZITER = 3 };

// C[M,d] = f_A(A)[M,d] @ B[d,d] with fused elementwise A-transform + epilogue.
// Block = 8 waves; the block stages f_A(A[m0:m0+64, :]) into LDS once
// (coalesced b128), then each wave computes a 64x32 output tile = 4x2 WMMA
// 16x16 subtiles with V_WMMA_F32_16X16X4_F32, A-fragments from LDS
// (conflict-free ds_load_b64 via stride d+4), B streamed from global
// (256 KB W/Wt stays hot in WGP$/L2 across all blocks).
template <int AM, int EM>
__global__ void __launch_bounds__(256)
gemm_wmma_k(const float* __restrict__ A,    // raw A / vh (AM_T) / a_vh (AM_U)
            const float* __restrict__ P1,   // gz  (AM_T)
            const float* __restrict__ P2,   // uz  (AM_T)
            const float* __restrict__ B,    // d x d (W or Wt), row-major
            const float* __restrict__ bias, // d   (EPI_BIAS / EPI_ZITER)
            const float* __restrict__ X1,   // aux1
            const float* __restrict__ X2,   // aux2
            const float* __restrict__ X3,   // aux3
            float* __restrict__ Out,
            int M, int d, float alpha, float gscale)
{
  extern __shared__ float As[];             // 64 * (d + LDPAD) floats
  const int ldst = d + LDPAD;
  const int tid  = threadIdx.x;
  const int lane = tid & 31;
  const int wave = tid >> 5;
  const int m0   = blockIdx.x * 64;         // block owns a 64-row band
  const int n0   = wave * 32;               // 8 waves cover N = 256

  // ---- stage f_A(A band) into LDS: 4 floats/thread/iter, coalesced ----
  const int iters = (64 * d) >> 10;         // 16 when d == 256
  for (int j = 0; j < iters; ++j) {
    const int flat = (j << 10) + (tid << 2);
    const int r = flat / d;
    const int c = flat - r * d;
    const size_t goff = (size_t)(m0 + r) * (size_t)d + (size_t)c;
    v4f val;
    if (AM == AM_RAW) {
      val = *(const v4f*)(A + goff);
    } else if (AM == AM_T) {                // T = gz*vh^2 - uz
      v4f x = *(const v4f*)(A + goff);
      v4f g = *(const v4f*)(P1 + goff);
      v4f u = *(const v4f*)(P2 + goff);
#pragma unroll
      for (int q = 0; q < 4; ++q) val[q] = fmaf(g[q] * x[q], x[q], -u[q]);
    } else {                                // U = sigmoid(a)/softplus(a)
      v4f x = *(const v4f*)(A + goff);
#pragma unroll
      for (int q = 0; q < 4; ++q) val[q] = sigmoid_f(x[q]) / softplus_f(x[q]);
    }
    *(v4f*)(&As[r * ldst + c]) = val;
  }
  __syncthreads();

  const int l15 = lane & 15;
  const int lh  = lane >> 4;                // 0/1

  v8f zero = {};
  v8f acc[4][2];
#pragma unroll
  for (int s = 0; s < 4; ++s)
#pragma unroll
    for (int t = 0; t < 2; ++t) acc[s][t] = zero;

  for (int k0 = 0; k0 < d; k0 += 4) {
    // A frag (m-subtile s): row = s*16+l15, cols k0+2*lh .. +1 (ds_load_b64)
    v2f afrag[4];
#pragma unroll
    for (int s = 0; s < 4; ++s)
      afrag[s] = *(const v2f*)(&As[(s * 16 + l15) * ldst + k0 + 2 * lh]);

    // B frag: n = n0+t*16+l15, rows k0+2*lh (vgpr0), +1 (vgpr1)
    v2f bfrag[2];
#pragma unroll
    for (int t = 0; t < 2; ++t) {
      const size_t bidx = (size_t)(k0 + 2 * lh) * (size_t)d +
                          (size_t)(n0 + t * 16 + l15);
      bfrag[t].x = B[bidx];
      bfrag[t].y = B[bidx + d];
    }
#pragma unroll
    for (int s = 0; s < 4; ++s)
#pragma unroll
      for (int t = 0; t < 2; ++t)
        acc[s][t] = __builtin_amdgcn_wmma_f32_16x16x4_f32(
            false, afrag[s], false, bfrag[t], (short)0, acc[s][t],
            false, false);
  }

  // Epilogue. C/D layout: VGPR r -> M = r + 8*lh, N = l15.
#pragma unroll
  for (int s = 0; s < 4; ++s) {
#pragma unroll
    for (int t = 0; t < 2; ++t) {
#pragma unroll
      for (int r = 0; r < 8; ++r) {
        const int row = m0 + s * 16 + r + 8 * lh;
        const int col = n0 + t * 16 + l15;
        const size_t idx = (size_t)row * (size_t)d + (size_t)col;
        const float a = acc[s][t][r];
        float o;
        if (EM == EPI_BIAS) {
          o = a + bias[col];
        } else if (EM == EPI_AXPY) {
          o = X1[idx] + alpha * a;                              // vh/v updates
        } else if (EM == EPI_ZD) {
          o = X1[idx] + gscale * X2[idx] * X3[idx] + alpha * a; // D
        } else { // EPI_ZITER: zn = D + (g/2)*softplus(zn@W+b)*vh
          o = X1[idx] + gscale * softplus_f(a + bias[col]) * X2[idx];
        }
        Out[idx] = o;
      }
    }
  }
}

__global__ void prep_k(const float* __restrict__ az, float* __restrict__ sz,
                       float* __restrict__ gz, float* __restrict__ uz,
                       long n) {
  long i = (long)blockIdx.x * blockDim.x + threadIdx.x;
  if (i < n) {
    float a = az[i];
    float s = softplus_f(a);
    float g = sigmoid_f(a);
    sz[i] = s;
    gz[i] = g;
    uz[i] = g / s;
  }
}

__global__ void transpose_k(const float* __restrict__ W,
                            float* __restrict__ Wt, int d) {
  int i = blockIdx.x * blockDim.x + threadIdx.x;
  if (i < d * d) {
    int r = i / d, c = i % d;
    Wt[(size_t)c * d + r] = W[i];
  }
}

extern "C" void kernel_launch(void* const* d_in, const int* in_sizes, int n_in,
                              void* d_out, int out_size, void* d_ws,
                              size_t ws_size, hipStream_t stream) {
  const float* z0   = (const float*)d_in[0];
  const float* v0   = (const float*)d_in[1];
  const float* W    = (const float*)d_in[2];
  const float* bias = (const float*)d_in[3];
  const int    d    = in_sizes[3];                 // 256
  const size_t bd   = (size_t)in_sizes[0];         // b*d
  const int    M    = (int)(bd / (size_t)d);       // 16384

  float* p    = (float*)d_ws;
  float* Wt   = p; p += (size_t)d * d;
  float* zbuf = p; p += bd;
  float* vbuf = p; p += bd;
  float* vhA  = p; p += bd;
  float* vhB  = p; p += bd;
  float* znA  = p; p += bd;
  float* az   = p; p += bd;   // also reused for a_vh mid-step
  float* sz   = p; p += bd;
  float* gz   = p; p += bd;
  float* uz   = p; p += bd;
  float* Dbuf = p; p += bd;
  float* znB  = vhA;          // vhA is free once the v-half loop finishes

  const int    ewBlocks = (int)((bd + 255) / 256);
  const int    tBlocks  = (d * d + 255) / 256;
  const int    gBlocks  = M / 64;                  // one 64-row band per block
  const size_t smem     = (size_t)64 * (size_t)(d + LDPAD) * sizeof(float);
  const float  hg = 0.5f * GAMMA;                  // gamma/2
  const float  qg = 0.25f * GAMMA;                 // gamma/4

  transpose_k<<<tBlocks, 256, 0, stream>>>(W, Wt, d);
  hipMemcpyAsync(zbuf, z0, bd * sizeof(float), hipMemcpyDeviceToDevice, stream);
  hipMemcpyAsync(vbuf, v0, bd * sizeof(float), hipMemcpyDeviceToDevice, stream);

  // a_z = z@W + bias ; sz/gz/uz
  gemm_wmma_k<AM_RAW, EPI_BIAS><<<gBlocks, 256, smem, stream>>>(
      zbuf, nullptr, nullptr, W, bias, nullptr, nullptr, nullptr, az, M, d, 0.f, 0.f);
  prep_k<<<ewBlocks, 256, 0, stream>>>(az, sz, gz, uz, (long)bd);

  for (int step = 0; step < LSTEPS; ++step) {
    // --- v-half fixed point: vh <- v - (g/4) * (gz*vh^2 - uz) @ Wt ---
    float* vh = vhA;
    for (int it = 0; it < NFX; ++it) {
      float* outv = (it & 1) ? vhB : vhA;
      const float* asrc = (it == 0) ? vbuf : ((it & 1) ? vhA : vhB);
      gemm_wmma_k<AM_T, EPI_AXPY><<<gBlocks, 256, smem, stream>>>(
          asrc, gz, uz, Wt, nullptr, vbuf, nullptr, nullptr, outv, M, d, -qg, 0.f);
      vh = outv;
    } // NFX even -> vh == vhB

    // --- a_vh = vh@W + bias (reuse az buffer) ---
    gemm_wmma_k<AM_RAW, EPI_BIAS><<<gBlocks, 256, smem, stream>>>(
        vh, nullptr, nullptr, W, bias, nullptr, nullptr, nullptr, az, M, d, 0.f, 0.f);

    // --- D = z + (g/2)*sz*vh - (g/2) * (sig(a_vh)/sp(a_vh)) @ Wt ---
    gemm_wmma_k<AM_U, EPI_ZD><<<gBlocks, 256, smem, stream>>>(
        az, nullptr, nullptr, Wt, nullptr, zbuf, sz, vh, Dbuf, M, d, -hg, hg);

    // --- z fixed point: zn <- D + (g/2)*softplus(zn@W+b)*vh ---
    float* znew = znA;
    for (int it = 0; it < NFX; ++it) {
      float* outz = (it & 1) ? znB : znA;
      const float* asrc = (it == 0) ? zbuf : ((it & 1) ? znA : znB);
      gemm_wmma_k<AM_RAW, EPI_ZITER><<<gBlocks, 256, smem, stream>>>(
          asrc, nullptr, nullptr, W, bias, Dbuf, vh, nullptr, outz, M, d, 0.f, hg);
      znew = outz;
    } // NFX even -> znew == znB (== vhA)

    // --- a_z(new) = znew@W + bias ; refresh sz/gz/uz (also next step) ---
    gemm_wmma_k<AM_RAW, EPI_BIAS><<<gBlocks, 256, smem, stream>>>(
        znew, nullptr, nullptr, W, bias, nullptr, nullptr, nullptr, az, M, d, 0.f, 0.f);
    prep_k<<<ewBlocks, 256, 0, stream>>>(az, sz, gz, uz, (long)bd);

    // --- v_new = vh - (g/4) * (gz*vh^2 - uz) @ Wt ---
    gemm_wmma_k<AM_T, EPI_AXPY><<<gBlocks, 256, smem, stream>>>(
        vh, gz, uz, Wt, nullptr, vh, nullptr, nullptr, vbuf, M, d, -qg, 0.f);

    // --- z = znew ---
    hipMemcpyAsync(zbuf, znew, bd * sizeof(float), hipMemcpyDeviceToDevice, stream);
  }

  hipMemcpyAsync(d_out, zbuf, bd * sizeof(float), hipMemcpyDeviceToDevice, stream);
  hipMemcpyAsync((float*)d_out + bd, vbuf, bd * sizeof(float),
                 hipMemcpyDeviceToDevice, stream);
}